// GATModel_52682068853202
// MI455X (gfx1250) — compile-verified
//
#include <hip/hip_runtime.h>
#include <hip/hip_bf16.h>

#define N_NODES 50000
#define N_EDGES 800000
#define N_TOT   (N_EDGES + N_NODES)   // edges + self loops
#define N_GRAPHS 64
#define HID 128
#define NEG_SLOPE 0.2f
#define LN_EPS 1e-5f

typedef __attribute__((ext_vector_type(2))) float v2f;
typedef __attribute__((ext_vector_type(8))) float v8f;

// ---------------------------------------------------------------------------
// GEMM: H[M,128] = A[M,K] @ W[K,128] using V_WMMA_F32_16X16X4_F32.
// One wave per 16x16 tile; block = 256 threads = 8 waves = the 8 N-tiles of
// one 16-row stripe. M = 50000 = 3125*16 exactly, K in {64,128}.
// Fragment layouts per CDNA5 ISA (32-bit A 16x4 / B 4x16 / C 16x16).
// ---------------------------------------------------------------------------
__global__ __launch_bounds__(256)
void gemm_wmma_f32(const float* __restrict__ A, const float* __restrict__ W,
                   float* __restrict__ H, int K) {
  const int lane = threadIdx.x & 31;
  const int wave = threadIdx.x >> 5;        // n-tile 0..7
  const int tm   = blockIdx.x;              // m-tile 0..3124
  const int half = lane >> 4;               // 0: lanes 0-15, 1: lanes 16-31
  const int l16  = lane & 15;

  const float* Arow = A + (size_t)(tm * 16 + l16) * K;   // A row for this lane
  const float* Bcol = W + wave * 16 + l16;               // B column for this lane

  v8f acc = {};
  for (int k0 = 0; k0 < K; k0 += 4) {
    const int ks = k0 + 2 * half;
    v2f a, b;
    a.x = Arow[ks];
    a.y = Arow[ks + 1];
    b.x = Bcol[(size_t)ks * HID];
    b.y = Bcol[(size_t)(ks + 1) * HID];
    acc = __builtin_amdgcn_wmma_f32_16x16x4_f32(
        /*neg_a=*/false, a, /*neg_b=*/false, b,
        /*c_mod=*/(short)0, acc, /*reuse_a=*/false, /*reuse_b=*/false);
  }
  float* Cout = H + (size_t)(tm * 16) * HID + wave * 16 + l16;
#pragma unroll
  for (int r = 0; r < 8; ++r)
    Cout[(size_t)(r + 8 * half) * HID] = acc[r];
}

// ---------------------------------------------------------------------------
// Per-node attention scalars: s = h . asrc, t = h . adst  (one wave per node)
// ---------------------------------------------------------------------------
__global__ __launch_bounds__(256)
void st_kernel(const float* __restrict__ H, const float* __restrict__ asrc,
               const float* __restrict__ adst, float* __restrict__ s,
               float* __restrict__ t) {
  const int node = blockIdx.x * (blockDim.x >> 5) + (threadIdx.x >> 5);
  if (node >= N_NODES) return;
  const int lane = threadIdx.x & 31;
  const float* hr = H + (size_t)node * HID;
  float ss = 0.f, tt = 0.f;
#pragma unroll
  for (int j = 0; j < 4; ++j) {
    const int f = lane + 32 * j;
    const float hv = hr[f];
    ss += hv * asrc[f];
    tt += hv * adst[f];
  }
#pragma unroll
  for (int m = 16; m > 0; m >>= 1) {
    ss += __shfl_xor(ss, m, 32);
    tt += __shfl_xor(tt, m, 32);
  }
  if (lane == 0) { s[node] = ss; t[node] = tt; }
}

// ---------------------------------------------------------------------------
// Per-layer init: zero aggregation accumulator, m = -inf, z = 0
// ---------------------------------------------------------------------------
__global__ void init_layer(float* __restrict__ O, float* __restrict__ m,
                           float* __restrict__ z) {
  const int i = blockIdx.x * blockDim.x + threadIdx.x;
  if (i < N_NODES * HID) O[i] = 0.f;
  if (i < N_NODES) { m[i] = -__builtin_inff(); z[i] = 0.f; }
}

__device__ __forceinline__ void atomicMaxFloat(float* addr, float value) {
  // init is -inf (0xFF800000): int-max for >=0, uint-min for <0 is consistent
  if (value >= 0.f)
    atomicMax((int*)addr, __float_as_int(value));
  else
    atomicMin((unsigned int*)addr, __float_as_uint(value));
}

// ---------------------------------------------------------------------------
// Edge pass 1: score = leaky_relu(s[src] + t[dst]); segment max into m[dst]
// ---------------------------------------------------------------------------
__global__ __launch_bounds__(256)
void edge_score(const int* __restrict__ ei, const float* __restrict__ s,
                const float* __restrict__ t, float* __restrict__ score,
                float* __restrict__ m) {
  const int e = blockIdx.x * blockDim.x + threadIdx.x;
  if (e >= N_TOT) return;
  int src, dst;
  if (e < N_EDGES) { src = ei[e]; dst = ei[N_EDGES + e]; }
  else             { src = dst = e - N_EDGES; }
  float sc = s[src] + t[dst];
  sc = sc > 0.f ? sc : NEG_SLOPE * sc;
  score[e] = sc;
  atomicMaxFloat(&m[dst], sc);
}

// ---------------------------------------------------------------------------
// Edge pass 2: e = exp(score - m[dst]); segment sum into z[dst]
// ---------------------------------------------------------------------------
__global__ __launch_bounds__(256)
void edge_exp(const int* __restrict__ ei, float* __restrict__ score,
              const float* __restrict__ m, float* __restrict__ z) {
  const int e = blockIdx.x * blockDim.x + threadIdx.x;
  if (e >= N_TOT) return;
  const int dst = (e < N_EDGES) ? ei[N_EDGES + e] : (e - N_EDGES);
  const float ev = __expf(score[e] - m[dst]);
  score[e] = ev;
  atomicAdd(&z[dst], ev);
}

// ---------------------------------------------------------------------------
// Edge pass 3: out[dst] += (e/z[dst]) * h[src]  (one wave per edge, 4 feat/lane)
// ---------------------------------------------------------------------------
__global__ __launch_bounds__(256)
void edge_scatter(const int* __restrict__ ei, const float* __restrict__ score,
                  const float* __restrict__ z, const float* __restrict__ H,
                  float* __restrict__ O) {
  const int e = blockIdx.x * (blockDim.x >> 5) + (threadIdx.x >> 5);
  if (e >= N_TOT) return;
  const int lane = threadIdx.x & 31;
  int src, dst;
  if (e < N_EDGES) { src = ei[e]; dst = ei[N_EDGES + e]; }
  else             { src = dst = e - N_EDGES; }
  const float alpha = score[e] / z[dst];
  const float* hs = H + (size_t)src * HID;
  float* od = O + (size_t)dst * HID;
#pragma unroll
  for (int j = 0; j < 4; ++j) {
    const int f = lane + 32 * j;
    atomicAdd(&od[f], alpha * hs[f]);
  }
}

// ---------------------------------------------------------------------------
// out = relu(LayerNorm(O + bias)) in place  (one wave per node)
// ---------------------------------------------------------------------------
__global__ __launch_bounds__(256)
void ln_relu(float* __restrict__ O, const float* __restrict__ bias,
             const float* __restrict__ g, const float* __restrict__ b) {
  const int node = blockIdx.x * (blockDim.x >> 5) + (threadIdx.x >> 5);
  if (node >= N_NODES) return;
  const int lane = threadIdx.x & 31;
  float* row = O + (size_t)node * HID;
  float v[4];
  float sum = 0.f;
#pragma unroll
  for (int j = 0; j < 4; ++j) {
    const int f = lane + 32 * j;
    v[j] = row[f] + bias[f];
    sum += v[j];
  }
#pragma unroll
  for (int m = 16; m > 0; m >>= 1) sum += __shfl_xor(sum, m, 32);
  const float mu = sum * (1.f / HID);
  float vs = 0.f;
#pragma unroll
  for (int j = 0; j < 4; ++j) { const float d = v[j] - mu; vs += d * d; }
#pragma unroll
  for (int m = 16; m > 0; m >>= 1) vs += __shfl_xor(vs, m, 32);
  const float rstd = rsqrtf(vs * (1.f / HID) + LN_EPS);
#pragma unroll
  for (int j = 0; j < 4; ++j) {
    const int f = lane + 32 * j;
    float o = (v[j] - mu) * rstd * g[f] + b[f];
    row[f] = o > 0.f ? o : 0.f;
  }
}

// ---------------------------------------------------------------------------
// Pooling
// ---------------------------------------------------------------------------
__global__ void zero_pool(float* __restrict__ pooled, float* __restrict__ cnt) {
  const int i = blockIdx.x * blockDim.x + threadIdx.x;
  if (i < N_GRAPHS * HID) pooled[i] = 0.f;
  if (i < N_GRAPHS) cnt[i] = 0.f;
}

__global__ __launch_bounds__(256)
void pool_kernel(const float* __restrict__ H, const int* __restrict__ batch,
                 float* __restrict__ pooled, float* __restrict__ cnt) {
  const int node = blockIdx.x * (blockDim.x >> 5) + (threadIdx.x >> 5);
  if (node >= N_NODES) return;
  const int lane = threadIdx.x & 31;
  const int gidx = batch[node];
  const float* hr = H + (size_t)node * HID;
#pragma unroll
  for (int j = 0; j < 4; ++j) {
    const int f = lane + 32 * j;
    atomicAdd(&pooled[gidx * HID + f], hr[f]);
  }
  if (lane == 0) atomicAdd(&cnt[gidx], 1.f);
}

// ---------------------------------------------------------------------------
// Head: z = relu(concat(pooled/cnt, gattr) @ fc1 + b1); out = z @ fc2 + b2
// One block (128 threads) per graph — tiny, latency-bound.
// ---------------------------------------------------------------------------
__global__ __launch_bounds__(128)
void head_kernel(const float* __restrict__ pooled, const float* __restrict__ cnt,
                 const float* __restrict__ gattr, const float* __restrict__ fc1w,
                 const float* __restrict__ fc1b, const float* __restrict__ fc2w,
                 const float* __restrict__ fc2b, float* __restrict__ out) {
  __shared__ float invec[HID + 8];
  __shared__ float red[HID];
  const int g = blockIdx.x, tid = threadIdx.x;
  float c = cnt[g];
  c = c > 1.f ? c : 1.f;
  invec[tid] = pooled[g * HID + tid] / c;
  if (tid < 8) invec[HID + tid] = gattr[g * 8 + tid];
  __syncthreads();
  float acc = fc1b[tid];
  for (int k = 0; k < HID + 8; ++k) acc += invec[k] * fc1w[k * HID + tid];
  const float zz = acc > 0.f ? acc : 0.f;
  red[tid] = zz * fc2w[tid];
  __syncthreads();
  for (int s = 64; s > 0; s >>= 1) {
    if (tid < s) red[tid] += red[tid + s];
    __syncthreads();
  }
  if (tid == 0) out[g] = red[0] + fc2b[0];
}

// ---------------------------------------------------------------------------
extern "C" void kernel_launch(void* const* d_in, const int* in_sizes, int n_in,
                              void* d_out, int out_size, void* d_ws, size_t ws_size,
                              hipStream_t stream) {
  const float* x     = (const float*)d_in[0];
  const int*   ei    = (const int*)d_in[1];
  const int*   batch = (const int*)d_in[2];
  const float* gattr = (const float*)d_in[3];
  const float* W[3]    = {(const float*)d_in[4],  (const float*)d_in[10], (const float*)d_in[16]};
  const float* asrc[3] = {(const float*)d_in[5],  (const float*)d_in[11], (const float*)d_in[17]};
  const float* adst[3] = {(const float*)d_in[6],  (const float*)d_in[12], (const float*)d_in[18]};
  const float* bgat[3] = {(const float*)d_in[7],  (const float*)d_in[13], (const float*)d_in[19]};
  const float* lng[3]  = {(const float*)d_in[8],  (const float*)d_in[14], (const float*)d_in[20]};
  const float* lnb[3]  = {(const float*)d_in[9],  (const float*)d_in[15], (const float*)d_in[21]};
  const float* fc1w = (const float*)d_in[22];
  const float* fc1b = (const float*)d_in[23];
  const float* fc2w = (const float*)d_in[24];
  const float* fc2b = (const float*)d_in[25];

  // workspace carve-out
  char* w = (char*)d_ws;
  auto carve = [&](size_t bytes) -> float* {
    float* p = (float*)w;
    w += (bytes + 255) & ~(size_t)255;
    return p;
  };
  const size_t NB = (size_t)N_NODES * HID * sizeof(float);
  float* B0    = carve(NB);
  float* B1    = carve(NB);
  float* B2    = carve(NB);
  float* sbuf  = carve((size_t)N_NODES * sizeof(float));
  float* tbuf  = carve((size_t)N_NODES * sizeof(float));
  float* mbuf  = carve((size_t)N_NODES * sizeof(float));
  float* zbuf  = carve((size_t)N_NODES * sizeof(float));
  float* score = carve((size_t)N_TOT * sizeof(float));
  float* pooled = carve((size_t)N_GRAPHS * HID * sizeof(float));
  float* cnt    = carve((size_t)N_GRAPHS * sizeof(float));

  const int gemm_grid  = N_NODES / 16;                // 3125
  const int node_grid  = (N_NODES + 7) / 8;           // wave-per-node kernels
  const int edge_grid  = (N_TOT + 255) / 256;
  const int edgew_grid = (N_TOT + 7) / 8;             // wave-per-edge
  const int init_grid  = (N_NODES * HID + 255) / 256;

  // layer rotation: L1: X=x(K=64) H=B0 O=B1 ; L2: X=B1 H=B2 O=B0 ; L3: X=B0 H=B1 O=B2
  const float* Xin[3] = {x, B1, B0};
  float*       Hb[3]  = {B0, B2, B1};
  float*       Ob[3]  = {B1, B0, B2};
  const int    Kd[3]  = {64, HID, HID};

  for (int L = 0; L < 3; ++L) {
    gemm_wmma_f32<<<gemm_grid, 256, 0, stream>>>(Xin[L], W[L], Hb[L], Kd[L]);
    st_kernel<<<node_grid, 256, 0, stream>>>(Hb[L], asrc[L], adst[L], sbuf, tbuf);
    init_layer<<<init_grid, 256, 0, stream>>>(Ob[L], mbuf, zbuf);
    edge_score<<<edge_grid, 256, 0, stream>>>(ei, sbuf, tbuf, score, mbuf);
    edge_exp<<<edge_grid, 256, 0, stream>>>(ei, score, mbuf, zbuf);
    edge_scatter<<<edgew_grid, 256, 0, stream>>>(ei, score, zbuf, Hb[L], Ob[L]);
    ln_relu<<<node_grid, 256, 0, stream>>>(Ob[L], bgat[L], lng[L], lnb[L]);
  }

  zero_pool<<<(N_GRAPHS * HID + 255) / 256, 256, 0, stream>>>(pooled, cnt);
  pool_kernel<<<node_grid, 256, 0, stream>>>(B2, batch, pooled, cnt);
  head_kernel<<<N_GRAPHS, HID, 0, stream>>>(pooled, cnt, gattr, fc1w, fc1b,
                                            fc2w, fc2b, (float*)d_out);
}